// SimDiff_26508538151740
// MI455X (gfx1250) — compile-verified
//
#include <hip/hip_runtime.h>
#include <hip/hip_bf16.h>
#include <stdint.h>

// Problem constants (fixed by the reference harness)
#define Q_LEN     2048
#define HIDDEN    4096
#define N_HEADS   32
#define IMG_START 64
#define IMG_LEN   1792
#define KEEP_K    1254            // round(1792 * (1 - 0.3))
#define N_KEEP    1510            // 64 + 1254 + (2048 - 64 - 1792)
#define N_ROWS    (N_HEADS * Q_LEN)   // 65536 reduction rows

// Stage-1 reduction geometry
#define ROW_CHUNKS     32         // gridDim.y
#define WAVES_PER_BLK  8          // 256 threads, wave32
#define TOTAL_WAVES    (ROW_CHUNKS * WAVES_PER_BLK)     // 256 partial rows
#define ROWS_PER_WAVE  (N_ROWS / TOTAL_WAVES)           // 256
#define WMMA_PER_WAVE  (ROWS_PER_WAVE / 4)              // 64

typedef float v2f __attribute__((ext_vector_type(2)));
typedef float v8f __attribute__((ext_vector_type(8)));

// ---------------------------------------------------------------------------
// Stage 1: column sums of attn (65536 x 2048) via V_WMMA_F32_16X16X4_F32 with
// A = ones(16x4).  D[m, n] = sum_k B[k, n] for every m, so any lane's c[0]
// holds the 4-row partial column sum; accumulate over 256 rows per wave.
// grid = (128 col tiles, 32 row chunks), block = 256 (8 waves).
// ---------------------------------------------------------------------------
__global__ void __launch_bounds__(256)
attn_colsum_wmma(const float* __restrict__ attn, float* __restrict__ partial) {
    const int lane = threadIdx.x & 31;
    const int wave = threadIdx.x >> 5;
    const int colBase = blockIdx.x << 4;                 // 16 cols per tile
    const int waveGlobal = blockIdx.y * WAVES_PER_BLK + wave;  // 0..255
    const int rowStart = waveGlobal * ROWS_PER_WAVE;
    const int half = lane >> 4;                          // 0: rows r,r+1   1: rows r+2,r+3
    const int col  = lane & 15;

    const float* __restrict__ p =
        attn + (size_t)(rowStart + half * 2) * Q_LEN + colBase + col;

    const v2f a = {1.0f, 1.0f};   // A-matrix: all ones
    v8f c = {};
#pragma unroll 4
    for (int it = 0; it < WMMA_PER_WAVE; ++it) {
        v2f b;
        b.x = p[0];
        b.y = p[Q_LEN];
        p += (size_t)4 * Q_LEN;
        // D = A(ones) x B + C  : accumulates 4-row column sums
        c = __builtin_amdgcn_wmma_f32_16x16x4_f32(
                /*neg_a=*/false, a, /*neg_b=*/false, b,
                /*c_mod=*/(short)0, c, /*reuse_a=*/false, /*reuse_b=*/false);
    }
    // Every output row of D is identical; lanes 0-15 hold N = 0..15 in c[0].
    if (lane < 16) {
        partial[(size_t)waveGlobal * Q_LEN + colBase + col] = c[0];
    }
}

// ---------------------------------------------------------------------------
// Stage 2: deterministic reduction of the 256 partial rows -> attn_sum[2048]
// (mean scaling omitted: it does not change the top-k selection)
// ---------------------------------------------------------------------------
__global__ void __launch_bounds__(256)
reduce_partials(const float* __restrict__ partial, float* __restrict__ attn_sum) {
    const int j = blockIdx.x * 256 + threadIdx.x;
    if (j >= Q_LEN) return;
    float s = 0.0f;
#pragma unroll 8
    for (int k = 0; k < TOTAL_WAVES; ++k) {
        s += partial[(size_t)k * Q_LEN + j];
    }
    attn_sum[j] = s;
}

// ---------------------------------------------------------------------------
// Select top-KEEP_K of the 1792 image-token scores (ties: lower index wins,
// matching lax.top_k), then build the sorted keep index list of length 1510.
// Single block, 256 threads, fully deterministic.
// ---------------------------------------------------------------------------
__global__ void __launch_bounds__(256)
select_and_scan(const float* __restrict__ attn_sum, int* __restrict__ keep_idx) {
    __shared__ float vals[IMG_LEN];
    __shared__ unsigned char flag[Q_LEN];
    __shared__ int cnt[256];
    __shared__ int off[256];

    const int tid = threadIdx.x;

    for (int i = tid; i < IMG_LEN; i += 256) vals[i] = attn_sum[IMG_START + i];
    __syncthreads();

    // Exact rank selection: rank(j) = #{v_i > v_j} + #{v_i == v_j, i < j}
    for (int j = tid; j < IMG_LEN; j += 256) {
        const float v = vals[j];
        int rank = 0;
        for (int i = 0; i < IMG_LEN; ++i) {
            const float u = vals[i];
            rank += (u > v) ? 1 : ((u == v && i < j) ? 1 : 0);
        }
        flag[IMG_START + j] = (rank < KEEP_K) ? 1 : 0;
    }
    // Always-kept positions (text tokens before/after the image span)
    for (int i = tid; i < Q_LEN; i += 256) {
        if (i < IMG_START || i >= IMG_START + IMG_LEN) flag[i] = 1;
    }
    __syncthreads();

    // Block-wide exclusive scan over 8-element chunks
    const int base = tid * 8;
    int c = 0;
#pragma unroll
    for (int k = 0; k < 8; ++k) c += flag[base + k];
    cnt[tid] = c;
    __syncthreads();
    if (tid == 0) {
        int acc = 0;
        for (int t = 0; t < 256; ++t) { off[t] = acc; acc += cnt[t]; }
    }
    __syncthreads();

    int o = off[tid];
#pragma unroll
    for (int k = 0; k < 8; ++k) {
        if (flag[base + k]) keep_idx[o++] = base + k;
    }
}

// ---------------------------------------------------------------------------
// Gather hidden_states and position_embeddings rows (float4 coalesced).
// grid = (1510, 2): y==0 -> hs, y==1 -> pe.
// ---------------------------------------------------------------------------
__global__ void __launch_bounds__(256)
gather_rows(const float* __restrict__ hs, const float* __restrict__ pe,
            const int* __restrict__ keep_idx, float* __restrict__ out) {
    const int r   = blockIdx.x;
    const int src = keep_idx[r];
    const float* __restrict__ in = (blockIdx.y == 0) ? hs : pe;
    const float4* __restrict__ s4 =
        (const float4*)(in + (size_t)src * HIDDEN);
    float4* __restrict__ d4 =
        (float4*)(out + (size_t)blockIdx.y * ((size_t)N_KEEP * HIDDEN) +
                  (size_t)r * HIDDEN);
#pragma unroll 2
    for (int i = threadIdx.x; i < HIDDEN / 4; i += 256) d4[i] = s4[i];
}

// ---------------------------------------------------------------------------
// Gather attention mask: out[r, c] = am[keep_idx[r], keep_idx[c]]
// ---------------------------------------------------------------------------
__global__ void __launch_bounds__(256)
gather_mask(const float* __restrict__ am, const int* __restrict__ keep_idx,
            float* __restrict__ out_am) {
    const int idx = blockIdx.x * 256 + threadIdx.x;
    if (idx >= N_KEEP * N_KEEP) return;
    const int r = idx / N_KEEP;
    const int c = idx - r * N_KEEP;
    out_am[idx] = am[(size_t)keep_idx[r] * Q_LEN + keep_idx[c]];
}

// ---------------------------------------------------------------------------
extern "C" void kernel_launch(void* const* d_in, const int* in_sizes, int n_in,
                              void* d_out, int out_size, void* d_ws, size_t ws_size,
                              hipStream_t stream) {
    const float* hs   = (const float*)d_in[0];   // (1, 2048, 4096)
    const float* pe   = (const float*)d_in[1];   // (1, 2048, 4096)
    const float* am   = (const float*)d_in[2];   // (1, 1, 2048, 2048)
    const float* attn = (const float*)d_in[3];   // (1, 32, 2048, 2048)
    (void)in_sizes; (void)n_in; (void)out_size; (void)ws_size;

    // Workspace layout
    float* partial  = (float*)d_ws;                                  // 256*2048 f32 = 2 MB
    float* attn_sum = partial + (size_t)TOTAL_WAVES * Q_LEN;         // 2048 f32
    int*   keep_idx = (int*)(attn_sum + Q_LEN);                      // 1510 i32

    // Output layout: hs | pe | am
    float* out_hs = (float*)d_out;
    float* out_am = out_hs + (size_t)2 * N_KEEP * HIDDEN;

    // 1) Column sums of attention via WMMA (ones-vector matmul)
    dim3 g1(Q_LEN / 16, ROW_CHUNKS);
    attn_colsum_wmma<<<g1, 256, 0, stream>>>(attn, partial);

    // 2) Deterministic final reduction
    reduce_partials<<<(Q_LEN + 255) / 256, 256, 0, stream>>>(partial, attn_sum);

    // 3) Top-k selection + sorted keep-index build
    select_and_scan<<<1, 256, 0, stream>>>(attn_sum, keep_idx);

    // 4) Row gathers for hidden_states / position_embeddings
    dim3 g4(N_KEEP, 2);
    gather_rows<<<g4, 256, 0, stream>>>(hs, pe, keep_idx, out_hs);

    // 5) 2-D gather of the attention mask
    const int nAm = N_KEEP * N_KEEP;
    gather_mask<<<(nAm + 255) / 256, 256, 0, stream>>>(am, keep_idx, out_am);
}